// Reprojection_22539988370062
// MI455X (gfx1250) — compile-verified
//
#include <hip/hip_runtime.h>

#define VOXX 240
#define VOXY 144
#define VOXZ 240
#define VOX_TOTAL (VOXX * VOXY * VOXZ)   // 8,294,400
#define IMG_H 480
#define IMG_W 640
#define HW (IMG_H * IMG_W)               // 307,200

typedef float float4v __attribute__((ext_vector_type(4)));

// ---------------------------------------------------------------------------
// Pattern fill via gfx1250 async LDS->global B128 stores (ASYNCcnt path).
// 1 KB LDS staging buffer holds the pattern; each lane streams 64 B per loop
// iteration as 4 x global_store_async_from_lds_b128 using IOFFSET 0/16/32/48.
// (ISA: INST_OFFSET is added to BOTH the global and LDS addresses, so the
// staging buffer covers lane*16 + 48 + 15 = 559 < 1024 bytes of pattern.)
// Hardware stalls issue when ASYNCcnt saturates; wait 0 before kernel end.
// ---------------------------------------------------------------------------
__global__ void fill_async_kernel(unsigned* __restrict__ out,
                                  long long n64,          // # of 64B units
                                  long long total_dwords, // total dwords
                                  unsigned pattern)
{
    __shared__ __align__(16) unsigned zbuf[256];  // 1 KB of pattern
    if (threadIdx.x < 256) zbuf[threadIdx.x] = pattern;
    __syncthreads();

    unsigned lds_off = (unsigned)(size_t)(&zbuf[(threadIdx.x & 31u) * 4u]);

    long long i      = (long long)blockIdx.x * blockDim.x + threadIdx.x;
    long long stride = (long long)gridDim.x * blockDim.x;

    for (; i < n64; i += stride) {
        const unsigned* g = out + (i << 4);
        asm volatile(
            "global_store_async_from_lds_b128 %0, %1, off\n\t"
            "global_store_async_from_lds_b128 %0, %1, off offset:16\n\t"
            "global_store_async_from_lds_b128 %0, %1, off offset:32\n\t"
            "global_store_async_from_lds_b128 %0, %1, off offset:48"
            :: "v"(g), "v"(lds_off)
            : "memory");
    }

#if __has_builtin(__builtin_amdgcn_s_wait_asynccnt)
    __builtin_amdgcn_s_wait_asynccnt(0);
#else
    asm volatile("s_wait_asynccnt 0x0" ::: "memory");
#endif

    if (blockIdx.x == 0 && threadIdx.x == 0) {
        for (long long k = (n64 << 4); k < total_dwords; ++k) out[k] = pattern;
    }
}

// ---------------------------------------------------------------------------
// Build inverse map: inv[idx] = pixel p (valid indices are unique, any-wins
// matches the reference's unordered .set semantics for the degenerate case).
// ---------------------------------------------------------------------------
__global__ void build_inv_kernel(const int* __restrict__ map,
                                 int* __restrict__ inv)
{
    int p = blockIdx.x * blockDim.x + threadIdx.x;
    int b = blockIdx.y;
    if (p >= HW) return;
    int idx = map[(long long)b * HW + p];
    if (idx > 0 && idx < VOX_TOTAL)
        inv[(long long)b * VOX_TOTAL + idx] = p;
}

// ---------------------------------------------------------------------------
// Gather pass: each thread owns 4 consecutive voxels. Reads int4 of inverse
// indices (coalesced b128), gathers up to 48 dwords from the ~20MB image
// (L2-resident), writes 12 float4 non-temporal coalesced stores. Every output
// element is written exactly once -> no zero-fill of the 398MB output needed.
// ---------------------------------------------------------------------------
__global__ void gather_out_kernel(const float* __restrict__ x,
                                  const int*   __restrict__ inv,
                                  float*       __restrict__ out,
                                  int C, int nc)
{
    long long v4  = (long long)blockIdx.x * blockDim.x + threadIdx.x;
    long long nv4 = (long long)VOX_TOTAL >> 2;
    if (v4 >= nv4) return;
    int b = blockIdx.y;

    const int4* invb = (const int4*)(inv + (long long)b * VOX_TOTAL);
    int4 p4 = invb[v4];

    const float* xb = x + ((long long)b * C + 1) * HW;   // skip channel 0
    float* ob = out + (long long)b * nc * VOX_TOTAL + (v4 << 2);

#pragma unroll
    for (int c = 0; c < 12; ++c) {
        if (c >= nc) break;
        const float* xc = xb + (long long)c * HW;
        float4v r;
        r.x = (p4.x >= 0) ? xc[p4.x] : 0.0f;
        r.y = (p4.y >= 0) ? xc[p4.y] : 0.0f;
        r.z = (p4.z >= 0) ? xc[p4.z] : 0.0f;
        r.w = (p4.w >= 0) ? xc[p4.w] : 0.0f;
        __builtin_nontemporal_store(r, (float4v*)(ob + (long long)c * VOX_TOTAL));
    }
}

// ---------------------------------------------------------------------------
// Fallback scatter (used only if d_ws is too small for the inverse map).
// ---------------------------------------------------------------------------
__global__ void reprojection_scatter_kernel(const float* __restrict__ x,
                                            const int*   __restrict__ map,
                                            float*       __restrict__ out,
                                            int C, int nc)
{
    int p = blockIdx.x * blockDim.x + threadIdx.x;
    int b = blockIdx.y;
    if (p >= HW) return;

    int idx = map[(long long)b * HW + p];
    if (idx <= 0 || idx >= VOX_TOTAL) return;

    const float* xb = x + ((long long)b * C + 1) * HW + p;
    float*       ob = out + (long long)b * nc * VOX_TOTAL + idx;

#pragma unroll
    for (int c = 0; c < 12; ++c) {
        if (c < nc) ob[(long long)c * VOX_TOTAL] = xb[(long long)c * HW];
    }
}

extern "C" void kernel_launch(void* const* d_in, const int* in_sizes, int n_in,
                              void* d_out, int out_size, void* d_ws, size_t ws_size,
                              hipStream_t stream)
{
    const float* x   = (const float*)d_in[0];   // (B, C, H, W) fp32
    const int*   map = (const int*)  d_in[1];   // (B, H*W) int32
    float*       out = (float*)d_out;           // (B, C-1, VOX_TOTAL) fp32

    int B = in_sizes[1] / HW;
    if (B < 1) B = 1;
    int C = in_sizes[0] / (B * HW);             // = 13
    int nc = C - 1;                             // = 12

    long long inv_dwords = (long long)B * VOX_TOTAL;
    long long need_bytes = inv_dwords * 4;      // ~33.2 MB for B=1

    if ((long long)ws_size >= need_bytes) {
        // --- Gather plan: fill inv=-1, invert map, single streaming pass ---
        int* inv = (int*)d_ws;

        fill_async_kernel<<<2048, 256, 0, stream>>>(
            (unsigned*)inv, inv_dwords >> 4, inv_dwords, 0xFFFFFFFFu);

        dim3 bgrid((HW + 255) / 256, B);
        build_inv_kernel<<<bgrid, 256, 0, stream>>>(map, inv);

        long long nv4 = (long long)VOX_TOTAL >> 2;
        dim3 ggrid((unsigned)((nv4 + 255) / 256), B);
        gather_out_kernel<<<ggrid, 256, 0, stream>>>(x, inv, out, C, nc);
    } else {
        // --- Fallback: zero-fill output + scatter -------------------------
        long long total = (long long)out_size;
        fill_async_kernel<<<4096, 256, 0, stream>>>(
            (unsigned*)out, total >> 4, total, 0u);

        dim3 sgrid((HW + 255) / 256, B);
        reprojection_scatter_kernel<<<sgrid, 256, 0, stream>>>(x, map, out, C, nc);
    }
}